// ConfidenceDrivenMaskLayer3D_85426899518020
// MI455X (gfx1250) — compile-verified
//
#include <hip/hip_runtime.h>
#include <math.h>

typedef __attribute__((ext_vector_type(2))) float    v2f;
typedef __attribute__((ext_vector_type(8))) float    v8f;
typedef __attribute__((ext_vector_type(4))) unsigned v4u;
typedef __attribute__((ext_vector_type(8))) unsigned v8u;

#define KSIZE   65      // Gaussian taps
#define KRAD    32
#define PATCH   80      // 16 outputs + 2*32 halo along conv axis
#define COLS    128     // columns per block (8 waves x 16)
#define PIT_RM  132     // row-major pitch (floats): patch[i][c]  (TDM pad 128+4)
#define PIT_TR  84      // transposed pitch (floats): patchT[c][i]
#define NSTEP   20      // 80 / 4 (K per V_WMMA_F32_16X16X4_F32)
#define NQUAD   (PATCH * COLS / 4)   // 2560 16-byte tasks per block

// ---- per-lane 16B global -> LDS copy, ASYNCcnt-tracked, no VGPR staging
__device__ __forceinline__ void async_copy16(unsigned lds_off, const void* gptr) {
    unsigned long long ga = (unsigned long long)(uintptr_t)gptr;
    asm volatile("global_load_async_to_lds_b128 %0, %1, off"
                 :: "v"(lds_off), "v"(ga) : "memory");
}
__device__ __forceinline__ void wait_async0() {
    asm volatile("s_wait_asynccnt 0x0" ::: "memory");
}

// ---- TDM: one descriptor moves a [nrows x 128 fp32] tile (line stride =
// strideElem) into LDS with a 4-dword pad after every 128 dwords -> pitch 132.
// D# layout per CDNA5 ISA ss8.3/8.4 (2D: groups 2/3 NULL).
__device__ __forceinline__ void tdm_load_2d(unsigned lds_off, const void* gptr,
                                            unsigned nrows,
                                            unsigned long long strideElem) {
    unsigned long long ga = (unsigned long long)(uintptr_t)gptr;
    const unsigned td0 = COLS;           // tensor/tile X length (no X OOB)
    v4u g0; v8u g1;
    g0[0] = 1u;                                        // count=1 (user D#)
    g0[1] = lds_off;                                   // lds_addr (bytes)
    g0[2] = (unsigned)ga;                              // global_addr[31:0]
    g0[3] = (unsigned)(ga >> 32) | (2u << 30);         // ga[56:32] | type=2
    g1[0] = (2u << 16)                                 // data_size = 4B
          | (1u << 20)                                 // pad_enable
          | (6u << 22)                                 // pad_interval: 128 dw
          | (3u << 25);                                // pad_amount : 4 dw
    g1[1] = (td0 & 0xFFFFu) << 16;                     // abar=0 | tdim0 lo16
    g1[2] = (td0 >> 16) | ((nrows & 0xFFFFu) << 16);   // tdim0 hi | tdim1 lo
    g1[3] = (nrows >> 16) | (td0 << 16);               // tdim1 hi | tile_dim0
    g1[4] = nrows & 0xFFFFu;                           // tile_dim1 | tile_dim2=0
    g1[5] = (unsigned)strideElem;                      // dim0_stride[31:0]
    g1[6] = (unsigned)(strideElem >> 32) & 0xFFFFu;    // dim0_stride[47:32]
    g1[7] = 0u;                                        // dim1_stride (2D: n/a)
    asm volatile("tensor_load_to_lds %0, %1" :: "s"(g0), "s"(g1) : "memory");
}

// Generic 1-D 65-tap Gaussian conv along one axis:
//   C[16x16] = Kband[16x80] x Data[80x16]   (20 x wmma_f32_16x16x4_f32)
// WMAJOR=false: axis strided, cols contiguous -> patch[i][c], TDM staging
// WMAJOR=true : axis contiguous (sA==1)       -> patchT[c][i], async staging
// mode: 0 = plain conv; 1 = conv*mask + (1-mask); 2 = conv*mask (final)
template <bool WMAJOR>
__global__ __launch_bounds__(256) void conv_pass(
    const float* __restrict__ in,
    const float* __restrict__ mask,
    float* __restrict__ out,
    int axisLen, int sA, int sB,
    int nAxisTiles, int nColTiles,
    int n2, int s1, int s2,
    int mode)
{
    __shared__ float ktab[KSIZE];
    __shared__ float patch[COLS * PIT_TR];   // 10752 >= 80*132 (10560)

    const int tid = threadIdx.x;

    // ---- Gaussian taps (x normalized by size, sigma = 1/40)
    if (tid < KSIZE) {
        float x = ((float)tid - (KSIZE - 1) * 0.5f) / (float)KSIZE;
        const float sig = 1.0f / 40.0f;
        ktab[tid] = __expf(-(x * x) / (2.0f * sig * sig));
    }
    __syncthreads();
    float ksum = 0.0f;
    for (int i = 0; i < KSIZE; ++i) ksum += ktab[i];   // redundant: EXEC uniform
    const float kinv = 1.0f / ksum;

    // ---- tile decomposition
    const int ntile = nAxisTiles * nColTiles;
    const int t  = blockIdx.x % ntile;
    const int o  = blockIdx.x / ntile;
    const int a0 = (t % nAxisTiles) * 16;
    const int b0 = (t / nAxisTiles) * COLS;
    const int baseO = (o / n2) * s1 + (o % n2) * s2;

    const int lane = tid & 31;
    const int wv   = tid >> 5;
    const int cb   = wv * 16;
    const int m    = lane & 15;
    const int hi   = lane >> 4;

    // prefetch the mask cacheline this lane needs at scatter time
    if (mode != 0)
        __builtin_prefetch(mask + baseO + (a0 + 8 * hi) * sA + (b0 + cb + m) * sB, 0, 1);

    const unsigned ldsbase = (unsigned)(uintptr_t)&patch[0];
    const float4 z4 = {};

    if (WMAJOR) {
        // ---- async-to-LDS staging: pack 4 along the contiguous axis (sA==1).
        // Tile origins are multiples of 16, KRAD=32 -> each quad wholly valid/OOB.
        for (int q = tid; q < NQUAD; q += 256) {
            int i0   = (q % (PATCH / 4)) * 4;
            int c0   = q / (PATCH / 4);
            int foff = c0 * PIT_TR + i0;
            int ap   = a0 - KRAD + i0;
            if (ap >= 0 && ap < axisLen)
                async_copy16(ldsbase + (unsigned)foff * 4u,
                             in + baseO + ap + (b0 + c0) * sB);
            else
                *(float4*)&patch[foff] = z4;
        }
        wait_async0();          // this wave's async LDS writes complete
    } else {
        // ---- TDM staging: zero-fill halo rows, DMA the valid row range.
        int r0 = max(0, KRAD - a0);
        int r1 = min(PATCH, axisLen + KRAD - a0);
        for (int q = tid; q < NQUAD; q += 256) {
            int i  = q / (COLS / 4);
            int c0 = (q % (COLS / 4)) * 4;
            if (i < r0 || i >= r1)
                *(float4*)&patch[i * PIT_RM + c0] = z4;
        }
        if (wv == 0) {          // wave-level DMA: EXEC ignored, one issuer
            tdm_load_2d(ldsbase + (unsigned)(r0 * PIT_RM) * 4u,
                        in + baseO + (a0 - KRAD + r0) * sA + b0,
                        (unsigned)(r1 - r0), (unsigned long long)sA);
            __builtin_amdgcn_s_wait_tensorcnt(0);
        }
    }
    __syncthreads();            // publish LDS across all 8 waves

    // ---- constant band matrix A[16x80], A[m][k] = ktap(k - m)
    // fp32 16x4 A layout: lanes0-15 K=4s,4s+1 ; lanes16-31 K=4s+2,4s+3
    v2f bandA[NSTEP];
    for (int s = 0; s < NSTEP; ++s) {
        int kk = 4 * s + 2 * hi;
        int t0 = kk - m, t1 = kk + 1 - m;
        bandA[s].x = (t0 >= 0 && t0 < KSIZE) ? ktab[t0] * kinv : 0.0f;
        bandA[s].y = (t1 >= 0 && t1 < KSIZE) ? ktab[t1] * kinv : 0.0f;
    }

    // ---- 20-step banded GEMM accumulation
    v8f acc = {};
    const int colw = cb + m;
    #pragma unroll
    for (int s = 0; s < NSTEP; ++s) {
        int i0 = 4 * s + 2 * hi;          // B halves mirror A halves
        v2f b;
        if (WMAJOR) {                     // consecutive dwords -> ds_load_b64
            b.x = patch[colw * PIT_TR + i0];
            b.y = patch[colw * PIT_TR + i0 + 1];
        } else {
            b.x = patch[i0 * PIT_RM + colw];
            b.y = patch[(i0 + 1) * PIT_RM + colw];
        }
        acc = __builtin_amdgcn_wmma_f32_16x16x4_f32(
                  false, bandA[s], false, b, (short)0, acc, false, false);
    }

    // ---- scatter C: VGPR r -> M=r (lanes 0-15) / M=r+8 (lanes 16-31)
    #pragma unroll
    for (int r = 0; r < 8; ++r) {
        int mrow = r + 8 * hi;
        int addr = baseO + (a0 + mrow) * sA + (b0 + cb + m) * sB;
        float v = acc[r];
        if (mode != 0) {
            float mv = mask[addr];
            v *= mv;                       // conf = blur * mask
            if (mode == 1) v += 1.0f - mv; // next init = conf + (1 - mask)
        }
        out[addr] = v;
    }
}

__global__ __launch_bounds__(256) void init_one_minus(
    const float* __restrict__ mask, float* __restrict__ out, int n)
{
    int i = blockIdx.x * blockDim.x + threadIdx.x;
    if (i < n) out[i] = 1.0f - mask[i];
}

extern "C" void kernel_launch(void* const* d_in, const int* in_sizes, int n_in,
                              void* d_out, int out_size, void* d_ws, size_t ws_size,
                              hipStream_t stream) {
    (void)n_in; (void)out_size; (void)ws_size;
    const float* mask = (const float*)d_in[0];
    float* out = (float*)d_out;

    const int N = in_sizes[0];              // 2*96*256*256 = 12,582,912
    float* buf0 = (float*)d_ws;             // init
    float* buf1 = buf0 + (size_t)N;         // after D pass
    float* buf2 = buf1 + (size_t)N;         // after H pass

    // strides (floats): W=1, H=256, D=256*256, batch=96*256*256
    const int sH = 256, sD = 65536, sBt = 6291456;

    init_one_minus<<<(N + 255) / 256, 256, 0, stream>>>(mask, buf0, N);

    for (int it = 0; it < 7; ++it) {
        // D-pass: axis D (96, stride 65536), cols along W; outer = B x H
        conv_pass<false><<<6 * 2 * (2 * 256), 256, 0, stream>>>(
            buf0, nullptr, buf1, 96, sD, 1, 6, 2, 256, sBt, sH, 0);
        // H-pass: axis H (256, stride 256), cols along W; outer = B x D
        conv_pass<false><<<16 * 2 * (2 * 96), 256, 0, stream>>>(
            buf1, nullptr, buf2, 256, sH, 1, 16, 2, 96, sBt, sD, 0);
        // W-pass: axis W (256, stride 1), cols along H; outer = B x D
        // fused: conf = blur*mask; init_next = conf + (1-mask); last -> conf
        int   mode = (it == 6) ? 2 : 1;
        float* dst = (it == 6) ? out : buf0;
        conv_pass<true><<<16 * 2 * (2 * 96), 256, 0, stream>>>(
            buf2, mask, dst, 256, 1, sH, 16, 2, 96, sBt, sD, mode);
    }
}